// KroneckerLayer_42099269435410
// MI455X (gfx1250) — compile-verified
//
#include <hip/hip_runtime.h>
#include <math.h>

// MI455X / gfx1250: wave32, WMMA f32 16x16x4.
typedef float v2f __attribute__((ext_vector_type(2)));
typedef float v8f __attribute__((ext_vector_type(8)));

#define XS 68   // row stride (dwords) for X / A^T / B^T in LDS  -> conflict-free b64 frag reads
#define TS 72   // row stride (dwords) for T in LDS              -> conflict-free b32 frag reads

__global__ __launch_bounds__(256) void kron_gelu_kernel(
    const float* __restrict__ x, const float* __restrict__ A,
    const float* __restrict__ B, const float* __restrict__ bias,
    float* __restrict__ out, int ntok)
{
  // XTs holds X_b (stride XS) during stage 1, then is reused for T (stride TS).
  __shared__ __align__(16) float XTs[64 * TS];   // 18432 B
  __shared__ __align__(16) float Ats[64 * XS];   // 17408 B : Ats[l*XS+k] = A[k][l]
  __shared__ __align__(16) float Bts[64 * XS];   // 17408 B : Bts[j*XS+i] = B[i][j]

  const int tid  = threadIdx.x;
  const int lane = tid & 31;
  const int w    = tid >> 5;      // wave 0..7
  const int half = lane >> 4;     // 0 / 1
  const int m    = lane & 15;

  // One-time: load A^T and B^T into LDS (transposed so fragment k-pairs are contiguous).
  for (int e = tid; e < 4096; e += 256) {
    int r = e >> 6, c = e & 63;
    Ats[c * XS + r] = A[e];       // A is [k][l] row-major
    Bts[c * XS + r] = B[e];       // B is [i][j] row-major
  }

  // Wave w owns C tiles (mt, nt) and (mt+2, nt): they share the B-fragment.
  const int mt = w >> 2;          // stage1: i-tile ; stage2: j-tile
  const int nt = w & 3;           // l-tile (both stages)

  // A-fragment (16x4, MxK): lane -> row m=lane&15, k = 4*kt + 2*half + v  (v = vgpr 0/1)
  const float* xr0 = &XTs[(16 * mt + m) * XS + 2 * half];
  const float* xr1 = xr0 + 32 * XS;
  // B-fragment (4x16, KxN) of A: lane -> col n=lane&15, k = 4*kt + 2*half + v
  const float* ar  = &Ats[(16 * nt + m) * XS + 2 * half];
  // Stage-2 A-fragment of B^T: element (j_local, i) = B[i][16*jt + j_local]
  const float* br0 = &Bts[(16 * mt + m) * XS + 2 * half];
  const float* br1 = br0 + 32 * XS;

  const v8f vzero = {0.f, 0.f, 0.f, 0.f, 0.f, 0.f, 0.f, 0.f};

  for (int b = blockIdx.x; b < ntok; b += gridDim.x) {
    __syncthreads();  // XTs free (also orders the one-time A/B loads on iter 0)

    // ---- stage 0: cooperatively stage X_b (64x64 f32) into LDS, b128 coalesced ----
    const float4* xg = (const float4*)(x + (size_t)b * 4096);
    #pragma unroll
    for (int r = 0; r < 4; ++r) {
      int f4 = r * 256 + tid;       // float4 index 0..1023
      int f  = f4 << 2;
      int i  = f >> 6, k = f & 63;  // 64 divisible by 4 -> float4 stays within a row
      *(float4*)&XTs[i * XS + k] = xg[f4];
    }
    __syncthreads();

    // ---- stage 1: T = X @ A  (two 16x16 tiles per wave, K = 64 in 16 steps of 4) ----
    v8f acc0 = vzero, acc1 = vzero;
    #pragma unroll
    for (int kt = 0; kt < 16; ++kt) {
      v2f a0 = *(const v2f*)(xr0 + 4 * kt);
      v2f a1 = *(const v2f*)(xr1 + 4 * kt);
      v2f bf = *(const v2f*)(ar  + 4 * kt);
      acc0 = __builtin_amdgcn_wmma_f32_16x16x4_f32(false, a0, false, bf, (short)0, acc0, false, false);
      acc1 = __builtin_amdgcn_wmma_f32_16x16x4_f32(false, a1, false, bf, (short)0, acc1, false, false);
    }
    __syncthreads();  // everyone done reading X before T overwrites the buffer

    // C layout: row = vgpr + 8*half, col = lane&15. Store T row-major [i][l], stride TS.
    {
      int l0 = 16 * nt + m;
      int i0 = 16 * mt + 8 * half;
      #pragma unroll
      for (int v = 0; v < 8; ++v) XTs[(i0 + v)      * TS + l0] = acc0[v];
      #pragma unroll
      for (int v = 0; v < 8; ++v) XTs[(i0 + 32 + v) * TS + l0] = acc1[v];
    }
    __syncthreads();

    // ---- stage 2: Y = B^T @ T, fused exact-GELU + bias, store to global ----
    v8f y0 = vzero, y1 = vzero;
    #pragma unroll
    for (int kt = 0; kt < 16; ++kt) {
      int kk = 4 * kt + 2 * half;
      v2f bf;
      bf.x = XTs[(kk    ) * TS + 16 * nt + m];
      bf.y = XTs[(kk + 1) * TS + 16 * nt + m];
      v2f a0 = *(const v2f*)(br0 + 4 * kt);
      v2f a1 = *(const v2f*)(br1 + 4 * kt);
      y0 = __builtin_amdgcn_wmma_f32_16x16x4_f32(false, a0, false, bf, (short)0, y0, false, false);
      y1 = __builtin_amdgcn_wmma_f32_16x16x4_f32(false, a1, false, bf, (short)0, y1, false, false);
    }

    float* og = out + (size_t)b * 4096;
    const int l0 = 16 * nt + m;
    #pragma unroll
    for (int v = 0; v < 8; ++v) {
      int j    = 16 * mt + 8 * half + v;
      int feat = (j << 6) + l0;                    // out feature = j*64 + l
      float val = y0[v];
      val = 0.5f * val * (1.0f + erff(val * 0.70710678118654752f));
      og[feat] = val + bias[feat];

      j += 32;
      feat = (j << 6) + l0;
      val = y1[v];
      val = 0.5f * val * (1.0f + erff(val * 0.70710678118654752f));
      og[feat] = val + bias[feat];
    }
    // loop-top __syncthreads protects XTs before the next token's X load
  }
}

extern "C" void kernel_launch(void* const* d_in, const int* in_sizes, int n_in,
                              void* d_out, int out_size, void* d_ws, size_t ws_size,
                              hipStream_t stream) {
  const float* x    = (const float*)d_in[0];
  const float* A    = (const float*)d_in[1];
  const float* B    = (const float*)d_in[2];
  const float* bias = (const float*)d_in[3];
  float* out = (float*)d_out;

  int ntok = in_sizes[0] / 4096;      // 16384
  int grid = 2048;                     // persistent-ish: A/B staged once per WG
  if (grid > ntok) grid = ntok;
  hipLaunchKernelGGL(kron_gelu_kernel, dim3(grid), dim3(256), 0, stream,
                     x, A, B, bias, out, ntok);
}